// loss_individual_36112085025035
// MI455X (gfx1250) — compile-verified
//
#include <hip/hip_runtime.h>
#include <hip/hip_bf16.h>

typedef __attribute__((ext_vector_type(2))) float v2f;
typedef __attribute__((ext_vector_type(8))) float v8f;

#define KC 20        // k-values staged in LDS per step (multiple of 4)
#define PLP 288      // pair-row stride (floats): 2*128 + 32; 288 % 64 == 32 -> lane halves on disjoint banks
#define GLP 136      // row stride for Cholesky matrix
#define SPLITK 25    // k-split chunks per class
#define MDIM 128
#define PDIM 32
#define REGEPS 1e-3f

// pair-interleaved LDS index: values for k and k+1 at the same column are adjacent (8B aligned)
#define PIDX(k, c) (((k) >> 1) * PLP + ((c) << 1) + ((k) & 1))

// ---------------------------------------------------------------------------
// 1) classify: U[j][k], j=0..3 circles (one-hot * w), j=4 "outside" * w
// ---------------------------------------------------------------------------
__global__ void cls_kernel(const float* __restrict__ x, const float* __restrict__ y,
                           const float* __restrict__ w, float* __restrict__ U, int K) {
    int k = blockIdx.x * 256 + threadIdx.x;
    if (k >= K) return;
    const float cx[4] = {0.5f, -0.5f, 0.5f, -0.5f};
    const float cy[4] = {0.5f, 0.5f, -0.5f, -0.5f};
    float xx = x[k], yy = y[k], wk = w[k];
    float u[5] = {0.f, 0.f, 0.f, 0.f, 0.f};
    bool any = false;
#pragma unroll
    for (int c = 0; c < 4; ++c) {
        float dx = xx - cx[c], dy = yy - cy[c];
        if (dx * dx + dy * dy <= 0.0625f) { u[c] = wk; any = true; }
    }
    if (!any) u[4] = wk;
#pragma unroll
    for (int j = 0; j < 5; ++j) U[j * K + k] = u[j];
}

// ---------------------------------------------------------------------------
// 2) gemm_base: Bpart[chunk][j][m][n] = sum_{k in chunk} U_j[k]*(dvX[k,m]duX[k,n]+dvY[k,m]duY[k,n])
//    grid = (SPLITK, 5); 8 waves in a 2x4 grid; each wave: 4 m-tiles x 2 n-tiles
// ---------------------------------------------------------------------------
__global__ __launch_bounds__(256) void gemm_base_kernel(
    const float* __restrict__ U, const float* __restrict__ dvX, const float* __restrict__ dvY,
    const float* __restrict__ duX, const float* __restrict__ duY,
    float* __restrict__ Bpart, int K, int CHUNK) {
    __shared__ float sAx[(KC / 2) * PLP];
    __shared__ float sAy[(KC / 2) * PLP];
    __shared__ float sBx[(KC / 2) * PLP];
    __shared__ float sBy[(KC / 2) * PLP];
    __shared__ float sU[KC];

    const int j = blockIdx.y, chunk = blockIdx.x;
    const int tid = threadIdx.x;
    const int wave = tid >> 5, lane = tid & 31, lanehi = lane >> 4, lanelo = lane & 15;
    const int wrow = wave >> 2;          // 0..1  -> m-tiles [wrow*4, wrow*4+4)
    const int wcol = wave & 3;           // 0..3  -> n-tiles [wcol*2, wcol*2+2)
    const int k0 = chunk * CHUNK;

    v8f acc[4][2];
    const v8f vzero = {0.f, 0.f, 0.f, 0.f, 0.f, 0.f, 0.f, 0.f};
#pragma unroll
    for (int a = 0; a < 4; ++a)
#pragma unroll
        for (int b = 0; b < 2; ++b) acc[a][b] = vzero;

    for (int s = 0; s < CHUNK; s += KC) {
        __syncthreads();
        if (tid < KC) sU[tid] = U[j * K + k0 + s + tid];
        __syncthreads();
        for (int idx = tid; idx < KC * (MDIM / 4); idx += 256) {
            int kk = idx >> 5, mc = (idx & 31) << 2;
            size_t goff = (size_t)(k0 + s + kk) * MDIM + mc;
            float uj = sU[kk];
            float4 dx = *(const float4*)(dvX + goff);
            float4 dy = *(const float4*)(dvY + goff);
            float4 ux = *(const float4*)(duX + goff);
            float4 uy = *(const float4*)(duY + goff);
            sAx[PIDX(kk, mc + 0)] = uj * dx.x;
            sAx[PIDX(kk, mc + 1)] = uj * dx.y;
            sAx[PIDX(kk, mc + 2)] = uj * dx.z;
            sAx[PIDX(kk, mc + 3)] = uj * dx.w;
            sAy[PIDX(kk, mc + 0)] = uj * dy.x;
            sAy[PIDX(kk, mc + 1)] = uj * dy.y;
            sAy[PIDX(kk, mc + 2)] = uj * dy.z;
            sAy[PIDX(kk, mc + 3)] = uj * dy.w;
            sBx[PIDX(kk, mc + 0)] = ux.x;
            sBx[PIDX(kk, mc + 1)] = ux.y;
            sBx[PIDX(kk, mc + 2)] = ux.z;
            sBx[PIDX(kk, mc + 3)] = ux.w;
            sBy[PIDX(kk, mc + 0)] = uy.x;
            sBy[PIDX(kk, mc + 1)] = uy.y;
            sBy[PIDX(kk, mc + 2)] = uy.z;
            sBy[PIDX(kk, mc + 3)] = uy.w;
        }
        __syncthreads();
#pragma unroll
        for (int kk = 0; kk < KC; kk += 4) {
            int row = ((kk >> 1) + lanehi) * PLP;     // pair-row for k = kk + 2*lanehi
            v2f ax[4], ay[4];
#pragma unroll
            for (int a = 0; a < 4; ++a) {
                int colA = (((wrow * 4 + a) * 16 + lanelo) << 1);
                ax[a] = *(const v2f*)&sAx[row + colA];
                ay[a] = *(const v2f*)&sAy[row + colA];
            }
#pragma unroll
            for (int b = 0; b < 2; ++b) {
                int colB = (((wcol * 2 + b) * 16 + lanelo) << 1);
                v2f bx = *(const v2f*)&sBx[row + colB];
                v2f by = *(const v2f*)&sBy[row + colB];
#pragma unroll
                for (int a = 0; a < 4; ++a) {
                    acc[a][b] = __builtin_amdgcn_wmma_f32_16x16x4_f32(false, ax[a], false, bx,
                                                                      (short)0, acc[a][b], false, false);
                    acc[a][b] = __builtin_amdgcn_wmma_f32_16x16x4_f32(false, ay[a], false, by,
                                                                      (short)0, acc[a][b], false, false);
                }
            }
        }
    }
    float* outp = Bpart + (size_t)(chunk * 5 + j) * (MDIM * MDIM);
#pragma unroll
    for (int a = 0; a < 4; ++a)
#pragma unroll
        for (int b = 0; b < 2; ++b)
#pragma unroll
            for (int e = 0; e < 8; ++e) {
                int mr = (wrow * 4 + a) * 16 + e + (lanehi ? 8 : 0);
                int nc = (wcol * 2 + b) * 16 + lanelo;
                outp[mr * MDIM + nc] = acc[a][b][e];
            }
}

// ---------------------------------------------------------------------------
// 3) reduce_parts: deterministic ordered sum over chunks -> Bbase[5][128][128]
// ---------------------------------------------------------------------------
__global__ void reduce_parts_kernel(const float* __restrict__ Bpart, float* __restrict__ Bbase) {
    int e = blockIdx.x * 256 + threadIdx.x;          // e < 5*16384
    int j = e >> 14, r = e & 16383;
    float s = 0.f;
    for (int c = 0; c < SPLITK; ++c) s += Bpart[(size_t)(c * 5 + j) * 16384 + r];
    Bbase[e] = s;
}

// ---------------------------------------------------------------------------
// 4) vec_kernel: wfv[m] = sum_k w f v[k,m];  svec[j][m] = sum_k U_j[k] dliftX[k] dvX[k,m]
// ---------------------------------------------------------------------------
__global__ void vec_kernel(const float* __restrict__ w, const float* __restrict__ f,
                           const float* __restrict__ v, const float* __restrict__ dliftX,
                           const float* __restrict__ dvX, const float* __restrict__ U,
                           float* __restrict__ wfv, float* __restrict__ svec, int K) {
    __shared__ float red[256];
    int m = blockIdx.x, tid = threadIdx.x;
    float acc[6] = {0.f, 0.f, 0.f, 0.f, 0.f, 0.f};
    for (int k = tid; k < K; k += 256) {
        float vm = v[k * MDIM + m];
        acc[0] += w[k] * f[k] * vm;
        float dl = dliftX[k] * dvX[k * MDIM + m];
#pragma unroll
        for (int j = 0; j < 5; ++j) acc[1 + j] += U[j * K + k] * dl;
    }
#pragma unroll
    for (int q = 0; q < 6; ++q) {
        red[tid] = acc[q];
        __syncthreads();
        for (int s = 128; s > 0; s >>= 1) {
            if (tid < s) red[tid] += red[tid + s];
            __syncthreads();
        }
        if (tid == 0) {
            if (q == 0) wfv[m] = red[0];
            else svec[(q - 1) * MDIM + m] = red[0];
        }
        __syncthreads();
    }
}

// ---------------------------------------------------------------------------
// 5) solve: per-p assemble B_p (pair-interleaved), G = B_p^T B_p + eps I (WMMA),
//    Cholesky, triangular solves, loss
// ---------------------------------------------------------------------------
__global__ __launch_bounds__(256) void solve_kernel(
    const float* __restrict__ sigma, const float* __restrict__ Bbase,
    const float* __restrict__ wfv, const float* __restrict__ svec, float* __restrict__ out) {
    extern __shared__ float sm[];
    float* Bp = sm;                        // (MDIM/2)*PLP  (pair-interleaved over m)
    float* Gs = Bp + (MDIM / 2) * PLP;     // MDIM*GLP
    float* lp = Gs + MDIM * GLP;           // 128
    float* rv = lp + MDIM;                 // 128
    float* yv = rv + MDIM;                 // 128
    float* Wv = yv + MDIM;                 // 128
    float* red = Wv + MDIM;                // 256

    const int p = blockIdx.x, tid = threadIdx.x;
    const float sg0 = sigma[p * 4 + 0], sg1 = sigma[p * 4 + 1];
    const float sg2 = sigma[p * 4 + 2], sg3 = sigma[p * 4 + 3];

    // assemble B_p (pair-interleaved over the m dimension) and l_p
    for (int i = tid; i < MDIM * MDIM; i += 256) {
        int m = i >> 7, n = i & 127;
        float b = Bbase[4 * 16384 + i] + sg0 * Bbase[i] + sg1 * Bbase[16384 + i] +
                  sg2 * Bbase[2 * 16384 + i] + sg3 * Bbase[3 * 16384 + i];
        Bp[PIDX(m, n)] = b;
    }
    if (tid < MDIM) {
        lp[tid] = wfv[tid] - (svec[4 * MDIM + tid] + sg0 * svec[tid] + sg1 * svec[MDIM + tid] +
                              sg2 * svec[2 * MDIM + tid] + sg3 * svec[3 * MDIM + tid]);
    }
    __syncthreads();

    // G = B_p^T B_p via WMMA: A[n, k=m] = Bp[m][n], B[k=m, n2] = Bp[m][n2]
    const int wave = tid >> 5, lane = tid & 31, lanehi = lane >> 4, lanelo = lane & 15;
    const int n0 = wave * 16;
    v8f g[8];
    const v8f vzero = {0.f, 0.f, 0.f, 0.f, 0.f, 0.f, 0.f, 0.f};
#pragma unroll
    for (int t = 0; t < 8; ++t) g[t] = vzero;
    for (int mm = 0; mm < MDIM; mm += 4) {
        int row = ((mm >> 1) + lanehi) * PLP;
        v2f a = *(const v2f*)&Bp[row + ((n0 + lanelo) << 1)];
#pragma unroll
        for (int t = 0; t < 8; ++t) {
            v2f b = *(const v2f*)&Bp[row + ((t * 16 + lanelo) << 1)];
            g[t] = __builtin_amdgcn_wmma_f32_16x16x4_f32(false, a, false, b,
                                                         (short)0, g[t], false, false);
        }
    }
#pragma unroll
    for (int t = 0; t < 8; ++t)
#pragma unroll
        for (int e = 0; e < 8; ++e)
            Gs[(n0 + e + (lanehi ? 8 : 0)) * GLP + t * 16 + lanelo] = g[t][e];
    __syncthreads();
    if (tid < MDIM) Gs[tid * GLP + tid] += REGEPS;     // + reg * I
    // rhs = B_p^T l_p
    if (tid < MDIM) {
        float r = 0.f;
        for (int m = 0; m < MDIM; ++m) r += Bp[PIDX(m, tid)] * lp[m];
        rv[tid] = r;
    }
    __syncthreads();

    // Cholesky (lower, in place in Gs)
    for (int jc = 0; jc < MDIM; ++jc) {
        if (tid == 0) Gs[jc * GLP + jc] = sqrtf(Gs[jc * GLP + jc]);
        __syncthreads();
        float djj = Gs[jc * GLP + jc];
        for (int i = jc + 1 + tid; i < MDIM; i += 256) Gs[i * GLP + jc] /= djj;
        __syncthreads();
        for (int i = jc + 1 + tid; i < MDIM; i += 256) {
            float lij = Gs[i * GLP + jc];
            for (int c = jc + 1; c <= i; ++c) Gs[i * GLP + c] -= lij * Gs[c * GLP + jc];
        }
        __syncthreads();
    }
    // forward: L y = rhs
    for (int i = 0; i < MDIM; ++i) {
        if (tid == 0) yv[i] = rv[i] / Gs[i * GLP + i];
        __syncthreads();
        float yi = yv[i];
        for (int k2 = i + 1 + tid; k2 < MDIM; k2 += 256) rv[k2] -= Gs[k2 * GLP + i] * yi;
        __syncthreads();
    }
    // backward: L^T W = y
    for (int i = MDIM - 1; i >= 0; --i) {
        if (tid == 0) Wv[i] = yv[i] / Gs[i * GLP + i];
        __syncthreads();
        float wi = Wv[i];
        for (int k2 = tid; k2 < i; k2 += 256) yv[k2] -= Gs[i * GLP + k2] * wi;
        __syncthreads();
    }
    // loss = || B_p W - l ||^2
    float part = 0.f;
    if (tid < MDIM) {
        float r = -lp[tid];
        for (int n = 0; n < MDIM; ++n) r += Bp[PIDX(tid, n)] * Wv[n];
        part = r * r;
    }
    red[tid] = part;
    __syncthreads();
    for (int s = 128; s > 0; s >>= 1) {
        if (tid < s) red[tid] += red[tid + s];
        __syncthreads();
    }
    if (tid == 0) out[p] = red[0];
}

// ---------------------------------------------------------------------------
extern "C" void kernel_launch(void* const* d_in, const int* in_sizes, int n_in,
                              void* d_out, int out_size, void* d_ws, size_t ws_size,
                              hipStream_t stream) {
    const float* sigma  = (const float*)d_in[0];
    const float* x      = (const float*)d_in[1];
    const float* y      = (const float*)d_in[2];
    const float* w      = (const float*)d_in[3];
    const float* f      = (const float*)d_in[4];
    const float* v      = (const float*)d_in[5];
    const float* dvX    = (const float*)d_in[6];
    const float* dvY    = (const float*)d_in[7];
    const float* dliftX = (const float*)d_in[8];
    const float* duX    = (const float*)d_in[9];
    const float* duY    = (const float*)d_in[10];

    const int K = in_sizes[1];           // 20000
    const int CHUNK = K / SPLITK;        // 800, multiple of KC

    float* ws    = (float*)d_ws;
    float* U     = ws;                                   // 5*K
    float* Bpart = U + (size_t)5 * K;                    // SPLITK*5*16384
    float* Bbase = Bpart + (size_t)SPLITK * 5 * 16384;   // 5*16384
    float* wfv   = Bbase + 5 * 16384;                    // 128
    float* svec  = wfv + MDIM;                           // 5*128

    cls_kernel<<<(K + 255) / 256, 256, 0, stream>>>(x, y, w, U, K);

    dim3 gB(SPLITK, 5);
    gemm_base_kernel<<<gB, 256, 0, stream>>>(U, dvX, dvY, duX, duY, Bpart, K, CHUNK);

    reduce_parts_kernel<<<(5 * 16384) / 256, 256, 0, stream>>>(Bpart, Bbase);

    vec_kernel<<<MDIM, 256, 0, stream>>>(w, f, v, dliftX, dvX, U, wfv, svec, K);

    size_t smemD = (size_t)((MDIM / 2) * PLP + MDIM * GLP + 4 * MDIM + 256) * sizeof(float);
    solve_kernel<<<PDIM, 256, smemD, stream>>>(sigma, Bbase, wfv, svec, (float*)d_out);
}